// GhebConv_V1_59863254171804
// MI455X (gfx1250) — compile-verified
//
#include <hip/hip_runtime.h>
#include <hip/hip_bf16.h>

typedef float v2f __attribute__((ext_vector_type(2)));
typedef float v8f __attribute__((ext_vector_type(8)));

#define N_NODES 50000
#define N_EDGES 600000
#define N_GRAPHS 500
#define D 128
#define D_OUT 32

#define M_TILE 80          // 50000 = 80 * 625 exactly -> no partial tiles
#define M_SUB  5           // 5 x 16-row subtiles per wave
#define LDS_PITCH 130      // bank-stagger (+2/row) and 8B alignment for even k

// ---------------- utility kernels ----------------

__global__ void zero_f32(float* __restrict__ p, int n) {
    int i = blockIdx.x * blockDim.x + threadIdx.x;
    if (i < n) p[i] = 0.0f;
}

__global__ void neg_copy(const float* __restrict__ src, float* __restrict__ dst, int n) {
    int i = blockIdx.x * blockDim.x + threadIdx.x;
    if (i < n) dst[i] = -src[i];
}

__global__ void count_deg(const int* __restrict__ src, float* __restrict__ deg, int nE) {
    int e = blockIdx.x * blockDim.x + threadIdx.x;
    if (e < nE) atomicAdd(&deg[src[e]], 1.0f);
}

__global__ void compute_dinv(const float* __restrict__ deg, float* __restrict__ dinv, int n) {
    int i = blockIdx.x * blockDim.x + threadIdx.x;
    if (i < n) {
        float d = deg[i];
        dinv[i] = (d > 0.0f) ? rsqrtf(d) : 0.0f;   // d >= 1 when > 0
    }
}

__global__ void compute_ew(const int* __restrict__ src, const int* __restrict__ dst,
                           const float* __restrict__ dinv, float* __restrict__ ew, int nE) {
    int e = blockIdx.x * blockDim.x + threadIdx.x;
    if (e < nE) ew[e] = -(dinv[src[e]] * dinv[dst[e]]);
}

// Repack W [3][128 k][128 n] so each lane's B fragment (k, k+1 at fixed n) is one b64:
// wp[ ((t*32 + k4)*2 + hi)*128 + n ] = { W[t][4*k4+2*hi][n], W[t][4*k4+2*hi+1][n] }
__global__ void pack_w(const float* __restrict__ W, float* __restrict__ wp) {
    int t = blockIdx.x * blockDim.x + threadIdx.x;
    if (t >= 3 * D * D) return;
    int mat = t >> 14, rem = t & 16383;
    int k = rem >> 7, n = rem & (D - 1);
    int pairIdx = (((mat << 5) | (k >> 2)) * 2 + ((k >> 1) & 1)) * D + n;
    wp[pairIdx * 2 + (k & 1)] = W[t];
}

// ---------------- SpMM scatter: Out[dst] += scale*ew*V[src] ----------------
__global__ __launch_bounds__(256)
void spmm_scatter(const int* __restrict__ src, const int* __restrict__ dst,
                  const float* __restrict__ ew, const float* __restrict__ V,
                  float* __restrict__ Out, float scale, int nE) {
    int t = blockIdx.x * blockDim.x + threadIdx.x;
    int e = t >> 5;
    if (e >= nE) return;
    int l = (t & 31) << 2;
    float w = scale * ew[e];
    const float* vin = V + (size_t)src[e] * D + l;
    float*       out = Out + (size_t)dst[e] * D + l;
    #pragma unroll
    for (int i = 0; i < 4; ++i)
        atomicAdd(out + i, w * vin[i]);
}

// ---------------- fused Chebyshev GEMM + bias + ReLU (V_WMMA_F32_16X16X4_F32) ----
// Q = relu(T0@W0 + T1@W1 + T2@W2 + b). Block = 256 thr (8 waves), 80-row M tile.
// Wave w owns N tile [16w,16w+16) with 5 M-subtile accumulators -> each B fragment
// feeds 5 wmma. A staged in LDS (8x wave reuse), W pre-packed for b64 loads.
__global__ __launch_bounds__(256)
void cheb_gemm_relu(const float* __restrict__ T0, const float* __restrict__ T1,
                    const float* __restrict__ T2, const float* __restrict__ Wp,
                    const float* __restrict__ bias, float* __restrict__ Q) {
    __shared__ float As[M_TILE][LDS_PITCH];
    const int wave  = threadIdx.x >> 5;
    const int lane  = threadIdx.x & 31;
    const int mBase = blockIdx.x * M_TILE;
    const int n0    = wave * 16;
    const int lo16  = lane & 15;
    const int hi    = lane >> 4;
    const int kOff  = hi * 2;
    const int n     = n0 + lo16;

    v8f acc[M_SUB];
    #pragma unroll
    for (int s = 0; s < M_SUB; ++s) acc[s] = (v8f){};

    const v2f* WpV = (const v2f*)Wp;

    #pragma unroll
    for (int t = 0; t < 3; ++t) {
        const float* T = (t == 0) ? T0 : (t == 1) ? T1 : T2;
        // cooperative A-tile load: 80 x 128 floats, 40 per thread
        #pragma unroll 8
        for (int i = 0; i < (M_TILE * D) / 256; ++i) {
            int idx = threadIdx.x + (i << 8);
            int r = idx >> 7, c = idx & (D - 1);
            As[r][c] = T[(size_t)(mBase + r) * D + c];
        }
        __syncthreads();

        #pragma unroll 4
        for (int ks = 0; ks < 32; ++ks) {            // k4 = ks*4
            v2f b = WpV[(((t << 5) | ks) * 2 + hi) * D + n];
            #pragma unroll
            for (int s = 0; s < M_SUB; ++s) {
                v2f a = *(const v2f*)&As[s * 16 + lo16][(ks << 2) + kOff];
                acc[s] = __builtin_amdgcn_wmma_f32_16x16x4_f32(
                    false, a, false, b, (short)0, acc[s], false, false);
            }
        }
        __syncthreads();
    }

    // C/D layout: VGPR v -> M = v + hi*8, N = lo16
    const float bv = bias[n];
    #pragma unroll
    for (int s = 0; s < M_SUB; ++s) {
        int mB = mBase + s * 16 + hi * 8;
        #pragma unroll
        for (int v = 0; v < 8; ++v) {
            float val = acc[s][v] + bv;
            Q[(size_t)(mB + v) * D + n] = val > 0.0f ? val : 0.0f;
        }
    }
}

// ---------------- pooling + output head ----------------

__global__ void pool_cnt(const int* __restrict__ batch, float* __restrict__ cnt, int n) {
    int i = blockIdx.x * blockDim.x + threadIdx.x;
    if (i < n) atomicAdd(&cnt[batch[i]], 1.0f);
}

__global__ void pool_sum(const float* __restrict__ H, const int* __restrict__ batch,
                         float* __restrict__ pooled, int n) {
    int t = blockIdx.x * blockDim.x + threadIdx.x;
    if (t >= n * D) return;
    int node = t >> 7, f = t & (D - 1);
    atomicAdd(&pooled[(size_t)batch[node] * D + f], H[(size_t)node * D + f]);
}

__global__ void head_gemm(const float* __restrict__ pooled, const float* __restrict__ cnt,
                          const float* __restrict__ Wout, const float* __restrict__ bout,
                          float* __restrict__ out) {
    int t = blockIdx.x * blockDim.x + threadIdx.x;
    if (t >= N_GRAPHS * D_OUT) return;
    int g = t / D_OUT, o = t % D_OUT;
    float inv = 1.0f / fmaxf(cnt[g], 1.0f);
    float acc = bout[o];
    #pragma unroll 4
    for (int f = 0; f < D; ++f)
        acc += pooled[(size_t)g * D + f] * inv * Wout[f * D_OUT + o];
    out[t] = acc;
}

// ---------------- host orchestration ----------------

extern "C" void kernel_launch(void* const* d_in, const int* in_sizes, int n_in,
                              void* d_out, int out_size, void* d_ws, size_t ws_size,
                              hipStream_t stream) {
    const float* x     = (const float*)d_in[0];
    const int*   src   = (const int*)d_in[1];
    const int*   dst   = ((const int*)d_in[1]) + N_EDGES;
    const int*   batch = (const int*)d_in[2];
    const float* W1 = (const float*)d_in[3];  const float* b1 = (const float*)d_in[4];
    const float* W2 = (const float*)d_in[5];  const float* b2 = (const float*)d_in[6];
    const float* W3 = (const float*)d_in[7];  const float* b3 = (const float*)d_in[8];
    const float* Wout = (const float*)d_in[9]; const float* bout = (const float*)d_in[10];
    float* out = (float*)d_out;

    float* ws = (float*)d_ws;
    const size_t HN = (size_t)N_NODES * D;
    const size_t WSZ = 3 * D * D;                 // 49152 floats per layer's W
    float* deg    = ws;
    float* dinv   = deg + N_NODES;
    float* ew     = dinv + N_NODES;
    float* bufA   = ew + N_EDGES;
    float* bufB   = bufA + HN;
    float* tx1    = bufB + HN;
    float* tx2    = tx1 + HN;
    float* wp1    = tx2 + HN;                     // packed weights, 3 layers
    float* wp2    = wp1 + WSZ;
    float* wp3    = wp2 + WSZ;
    float* pooled = wp3 + WSZ;
    float* cnt    = pooled + (size_t)N_GRAPHS * D;

    const int TPB = 256;
    auto blocks = [](long long nn, int tpb) { return (int)((nn + tpb - 1) / tpb); };

    // degrees + edge weights (shared across layers) and weight packing
    zero_f32<<<blocks(N_NODES, TPB), TPB, 0, stream>>>(deg, N_NODES);
    count_deg<<<blocks(N_EDGES, TPB), TPB, 0, stream>>>(src, deg, N_EDGES);
    compute_dinv<<<blocks(N_NODES, TPB), TPB, 0, stream>>>(deg, dinv, N_NODES);
    compute_ew<<<blocks(N_EDGES, TPB), TPB, 0, stream>>>(src, dst, dinv, ew, N_EDGES);
    pack_w<<<blocks(WSZ, TPB), TPB, 0, stream>>>(W1, wp1);
    pack_w<<<blocks(WSZ, TPB), TPB, 0, stream>>>(W2, wp2);
    pack_w<<<blocks(WSZ, TPB), TPB, 0, stream>>>(W3, wp3);

    const long long spmmThreads = (long long)N_EDGES * 32;
    const int gemmBlocks = N_NODES / M_TILE;      // 625 exact tiles

    struct Layer { const float* P; const float* Wp; const float* b; float* Q; };
    Layer layers[3] = { { x,    wp1, b1, bufA },
                        { bufA, wp2, b2, bufB },
                        { bufB, wp3, b3, bufA } };

    for (int l = 0; l < 3; ++l) {
        const Layer& L = layers[l];
        zero_f32<<<blocks((long long)HN, TPB), TPB, 0, stream>>>(tx1, (int)HN);
        spmm_scatter<<<blocks(spmmThreads, TPB), TPB, 0, stream>>>(
            src, dst, ew, L.P, tx1, 1.0f, N_EDGES);
        neg_copy<<<blocks((long long)HN, TPB), TPB, 0, stream>>>(L.P, tx2, (int)HN);
        spmm_scatter<<<blocks(spmmThreads, TPB), TPB, 0, stream>>>(
            src, dst, ew, tx1, tx2, 2.0f, N_EDGES);
        cheb_gemm_relu<<<gemmBlocks, TPB, 0, stream>>>(L.P, tx1, tx2, L.Wp, L.b, L.Q);
    }

    zero_f32<<<blocks(N_GRAPHS * D + N_GRAPHS, TPB), TPB, 0, stream>>>(
        pooled, N_GRAPHS * D + N_GRAPHS);
    pool_cnt<<<blocks(N_NODES, TPB), TPB, 0, stream>>>(batch, cnt, N_NODES);
    pool_sum<<<blocks((long long)N_NODES * D, TPB), TPB, 0, stream>>>(bufA, batch, pooled, N_NODES);
    head_gemm<<<blocks(N_GRAPHS * D_OUT, TPB), TPB, 0, stream>>>(pooled, cnt, Wout, bout, out);
}